// GraphAggregation_spatial_74285754351630
// MI455X (gfx1250) — compile-verified
//
#include <hip/hip_runtime.h>
#include <stdint.h>

// GraphAggregation spatial gather+transpose for MI455X (gfx1250).
// out[j, c, q] = yd[idx[q, j], c], flat (K=8, F=512, M=8192), fp32.
// Pure data movement: ~145 MB traffic floor -> ~6.2 us @ 23.3 TB/s.
// Uses CDNA5 async global->LDS gather (ASYNCcnt) to do the transpose in LDS,
// then 128B-coalesced non-temporal b128 stores.

namespace {
constexpr int Mdim = 8192;   // m (q dimension, innermost in output)
constexpr int Fdim = 512;    // f (columns of yd)
constexpr int Knb  = 8;      // neighbors
constexpr int QT   = 32;     // q-tile per workgroup (one wave-width)
constexpr int FC   = 256;    // columns per LDS pass (2 passes cover Fdim)
constexpr int LSTR = 36;     // LDS dword stride per c-row: 16B-aligned rows,
                             // lane stride 36 mod 64 banks -> <=2-way conflicts
} // namespace

typedef float v4f __attribute__((ext_vector_type(4)));   // native vector: OK for
                                                         // __builtin_nontemporal_*

__global__ __launch_bounds__(256)
void gather_tr_kernel(const float* __restrict__ yd,
                      const int*   __restrict__ idx,
                      float*       __restrict__ out)
{
    __shared__ __align__(16) float tile[FC * LSTR];   // 36 KB

    const int j    = blockIdx.y;              // neighbor index (outermost out dim)
    const int q0   = blockIdx.x * QT;         // q-tile base
    const int wave = (int)(threadIdx.x >> 5); // 0..7 (wave32)
    const int lane = (int)(threadIdx.x & 31);

    // Workgroup-relative LDS byte offset (low 32 bits of the flat LDS address).
    const uint32_t ldsbase = (uint32_t)(uintptr_t)(void*)&tile[0];

    // Each wave owns 4 of the 32 q-rows of this tile.
    int rows[4];
#pragma unroll
    for (int rr = 0; rr < 4; ++rr)
        rows[rr] = idx[(q0 + wave + 8 * rr) * Knb + j];

#pragma unroll
    for (int half = 0; half < 2; ++half) {
        const int cbase = half * FC;

        // ---- Phase 1: async gather rows -> transposed LDS tile[c][q] ----
        // Lanes read 32 consecutive columns of one yd row (128B coalesced);
        // the async engine scatters each lane's dword to tile[c][q] (b32
        // granularity is what makes the in-flight transpose possible).
#pragma unroll
        for (int rr = 0; rr < 4; ++rr) {
            const int ql = wave + 8 * rr;                       // 0..31
            const float* g = yd + (size_t)rows[rr] * Fdim + cbase + lane;
            uint32_t lds = ldsbase + (((uint32_t)lane * LSTR + (uint32_t)ql) << 2);
#pragma unroll
            for (int cc = 0; cc < FC; cc += 32) {
                asm volatile("global_load_async_to_lds_b32 %0, %1, off"
                             :: "v"(lds), "v"(g)
                             : "memory");
                g   += 32;
                lds += 32u * LSTR * 4u;
            }
        }

        // CDNA5 split counter for the async engine, then workgroup barrier so
        // every wave's LDS writes are visible to every reader.
        asm volatile("s_wait_asynccnt 0x0" ::: "memory");
        __syncthreads();

        // ---- Phase 2: b128 LDS reads along q, 128B-coalesced NT stores ----
        // lanes 8k..8k+7 handle column c+k; 8 lanes x float4 = 32 consecutive
        // q = one contiguous 128B store segment per column.
        const int sub = lane >> 3;   // 0..3 -> column offset within wave
        const int lo  = lane & 7;    // 0..7 -> q-quad within the 32-q tile
#pragma unroll
        for (int i = 0; i < FC / 32; ++i) {
            const int cl = wave * 4 + sub + 32 * i;   // 0..FC-1, each once per block
            const int c  = cbase + cl;
            const v4f v =
                *(const v4f*)&tile[(uint32_t)cl * LSTR + 4u * (uint32_t)lo];
            v4f* dst = (v4f*)(out + (size_t)(j * Fdim + c) * Mdim
                                  + (size_t)q0 + 4 * lo);
            // Output is written once and never re-read: non-temporal keeps the
            // L2 free for the fully-resident 16MB gather source.
            __builtin_nontemporal_store(v, dst);
        }
        __syncthreads();   // protect LDS before next half's async writes
    }
}

extern "C" void kernel_launch(void* const* d_in, const int* in_sizes, int n_in,
                              void* d_out, int out_size, void* d_ws, size_t ws_size,
                              hipStream_t stream) {
    (void)in_sizes; (void)n_in; (void)out_size; (void)d_ws; (void)ws_size;
    // d_in[0] = y_patch (m,64)  -- only its leading dim matters, unused here
    // d_in[1] = yd_patch (m,f) float32
    // d_in[2] = idx_k (1,m,K) int32 (harness integer convention)
    const float* yd  = (const float*)d_in[1];
    const int*   idx = (const int*)d_in[2];
    float*       out = (float*)d_out;

    dim3 grid(Mdim / QT, Knb, 1);   // 256 x 8 = 2048 workgroups, 16K waves
    gather_tr_kernel<<<grid, dim3(256, 1, 1), 0, stream>>>(yd, idx, out);
}